// DRNTRLLM_19971597926799
// MI455X (gfx1250) — compile-verified
//
#include <hip/hip_runtime.h>

typedef float v2f __attribute__((ext_vector_type(2)));
typedef float v8f __attribute__((ext_vector_type(8)));

#define D 128
#define NH 8
#define FOUT 16
#define NEG_SLOPE 0.2f

// ---- order-preserving float <-> uint key (for atomicMax on floats) ----
__device__ __forceinline__ unsigned f32_key(float f) {
    unsigned b = __float_as_uint(f);
    return (b & 0x80000000u) ? ~b : (b | 0x80000000u);
}
__device__ __forceinline__ float key_f32(unsigned k) {
    unsigned b = (k & 0x80000000u) ? (k & 0x7FFFFFFFu) : ~k;
    return __uint_as_float(b);
}

__device__ __forceinline__ float lrelu(float x) {
    return x > 0.0f ? x : NEG_SLOPE * x;
}

// XOR-butterfly add via ds_swizzle_b32 (group-of-32: and=0x1F, or=0, xor=mask).
// Masks < 16 stay within each 16-lane half -> per-half all-reduce.
#define SWIZZLE_ADD(t, mask)                                                              \
    t += __int_as_float(__builtin_amdgcn_ds_swizzle(__float_as_int(t), ((mask) << 10) | 0x1F))

// ---- zero out / denom, init global-max key ----
__global__ void gat_init_kernel(float* __restrict__ out, float* __restrict__ denom,
                                unsigned* __restrict__ gmaxk, int n_out, int n_den) {
    int i = blockIdx.x * blockDim.x + threadIdx.x;
    if (i < n_out) out[i] = 0.0f;
    if (i < n_den) denom[i] = 0.0f;
    if (i == 0) *gmaxk = f32_key(-__builtin_inff());
}

// ---- wt[h][k] = sum_f W[h*FOUT+f][k] * a[h][f]  (collapses trg GEMM to GEMV) ----
__global__ void gat_combine_a_kernel(const float* __restrict__ W, const float* __restrict__ a,
                                     float* __restrict__ wt) {
    int i = blockIdx.x * blockDim.x + threadIdx.x;
    if (i >= NH * D) return;
    int h = i / D, k = i % D;
    float s = 0.0f;
#pragma unroll
    for (int f = 0; f < FOUT; ++f)
        s = fmaf(W[(h * FOUT + f) * D + k], a[h * FOUT + f], s);
    wt[h * D + k] = s;
}

// ---- src projection (fp32 WMMA 16x16x4) + fused per-head logits s_src ----
// block = 256 threads (8 waves); wave w computes 16 nodes x head w (cols 16w..16w+15)
#define AT_STRIDE 132  // pad 128 -> 132 floats to avoid 16-way LDS bank conflicts
__global__ __launch_bounds__(256) void gat_src_proj_kernel(
    const float* __restrict__ X, const float* __restrict__ W, const float* __restrict__ a,
    float* __restrict__ proj, float* __restrict__ s, int N) {
    __shared__ float Atile[16 * AT_STRIDE];

    const int tid  = threadIdx.x;
    const int base = blockIdx.x * 16;

    // stage 16x128 node tile into LDS (coalesced, 8 floats/thread)
#pragma unroll
    for (int i = 0; i < 8; ++i) {
        int idx = tid + i * 256;
        int row = idx >> 7, col = idx & 127;
        int node = base + row;
        Atile[row * AT_STRIDE + col] = (node < N) ? X[(size_t)node * D + col] : 0.0f;
    }
    __syncthreads();

    const int wave = tid >> 5;       // head index (0..7), 16-column C tile
    const int lane = tid & 31;
    const int half = lane >> 4;      // 0 or 1
    const int l15  = lane & 15;
    const int kb   = half * 2;       // K sub-offset per ISA A/B 16x4 layout

    const float* Wrow = W + (size_t)(wave * 16 + l15) * D;  // W is (out=128, in=128)

    v8f c = {};
#pragma unroll 4
    for (int k = 0; k < D; k += 4) {
        v2f af, bf;
        const float* ap = &Atile[l15 * AT_STRIDE + k + kb];
        af.x = ap[0];
        af.y = ap[1];
        bf.x = Wrow[k + kb];         // B[k][n] = W[n][k]
        bf.y = Wrow[k + kb + 1];
        c = __builtin_amdgcn_wmma_f32_16x16x4_f32(false, af, false, bf,
                                                  (short)0, c, false, false);
    }

    // C layout: VGPR i, lanes 0-15 -> M=i, lanes 16-31 -> M=8+i; N = lane&15
    const float aw = a[wave * FOUT + l15];
#pragma unroll
    for (int i = 0; i < 8; ++i) {
        int m    = i + half * 8;
        int node = base + m;
        float v  = c[i];
        if (node < N) proj[(size_t)node * D + wave * 16 + l15] = v;
        // per-row dot with a[head]: all-reduce across each 16-lane half
        float t = v * aw;
        SWIZZLE_ADD(t, 1);
        SWIZZLE_ADD(t, 2);
        SWIZZLE_ADD(t, 4);
        SWIZZLE_ADD(t, 8);
        if (l15 == 0 && node < N) s[(size_t)node * NH + wave] = t;
    }
}

// ---- s_trg[n][h] = dot(trg[n,:], wt[h,:]) ----
__global__ void gat_trg_logits_kernel(const float* __restrict__ X, const float* __restrict__ wt,
                                      float* __restrict__ s, int N) {
    int i = blockIdx.x * blockDim.x + threadIdx.x;
    if (i >= N * NH) return;
    int n = i >> 3, h = i & 7;
    const float* x = X + (size_t)n * D;
    const float* w = wt + h * D;
    float acc = 0.0f;
#pragma unroll 8
    for (int k = 0; k < D; ++k) acc = fmaf(x[k], w[k], acc);
    s[i] = acc;
}

// ---- global max over all (edge, head) of lrelu(s_src[si]+s_trg[ti]) ----
__global__ __launch_bounds__(256) void gat_edge_max_kernel(
    const int* __restrict__ ei, const float* __restrict__ ssrc,
    const float* __restrict__ strg, unsigned* __restrict__ gmaxk, int E) {
    __shared__ unsigned red[256];
    int e = blockIdx.x * blockDim.x + threadIdx.x;
    float m = -__builtin_inff();
    if (e < E) {
        int si = ei[e], ti = ei[E + e];
        const float4* a4 = (const float4*)(ssrc + (size_t)si * NH);  // 32B-aligned rows
        const float4* b4 = (const float4*)(strg + (size_t)ti * NH);
        float4 a0 = a4[0], a1 = a4[1], b0 = b4[0], b1 = b4[1];
        m = fmaxf(m, a0.x + b0.x);  m = fmaxf(m, a0.y + b0.y);
        m = fmaxf(m, a0.z + b0.z);  m = fmaxf(m, a0.w + b0.w);
        m = fmaxf(m, a1.x + b1.x);  m = fmaxf(m, a1.y + b1.y);
        m = fmaxf(m, a1.z + b1.z);  m = fmaxf(m, a1.w + b1.w);
        m = lrelu(m);  // lrelu monotone: max(lrelu(x)) = lrelu(max(x))
    }
    red[threadIdx.x] = f32_key(m);
    __syncthreads();
    for (int off = 128; off > 0; off >>= 1) {
        if (threadIdx.x < off) {
            unsigned o = red[threadIdx.x + off];
            if (o > red[threadIdx.x]) red[threadIdx.x] = o;
        }
        __syncthreads();
    }
    if (threadIdx.x == 0) atomicMax(gmaxk, red[0]);
}

// ---- denom[ti][h] += exp(lrelu(score) - gmax) ----
__global__ void gat_edge_denom_kernel(const int* __restrict__ ei, const float* __restrict__ ssrc,
                                      const float* __restrict__ strg,
                                      const unsigned* __restrict__ gmaxk,
                                      float* __restrict__ denom, int E) {
    int e = blockIdx.x * blockDim.x + threadIdx.x;
    if (e >= E) return;
    float gm = key_f32(*gmaxk);
    int si = ei[e], ti = ei[E + e];
    const float4* a4 = (const float4*)(ssrc + (size_t)si * NH);
    const float4* b4 = (const float4*)(strg + (size_t)ti * NH);
    float4 a0 = a4[0], a1 = a4[1], b0 = b4[0], b1 = b4[1];
    float x[NH] = {a0.x + b0.x, a0.y + b0.y, a0.z + b0.z, a0.w + b0.w,
                   a1.x + b1.x, a1.y + b1.y, a1.z + b1.z, a1.w + b1.w};
    float* dst = denom + (size_t)ti * NH;
#pragma unroll
    for (int h = 0; h < NH; ++h)
        atomicAdd(&dst[h], expf(lrelu(x[h]) - gm));
}

// ---- out[ti][t] += src_proj[si][t] * attn[e][t>>4]; 2 edges/block, 1 feature/thread ----
__global__ __launch_bounds__(256) void gat_edge_aggregate_kernel(
    const int* __restrict__ ei, const float* __restrict__ ssrc, const float* __restrict__ strg,
    const unsigned* __restrict__ gmaxk, const float* __restrict__ denom,
    const float* __restrict__ proj, float* __restrict__ out, int E) {
    int e = blockIdx.x * 2 + (threadIdx.x >> 7);
    if (e >= E) return;
    int t = threadIdx.x & 127;
    int h = t >> 4;
    float gm = key_f32(*gmaxk);
    int si = ei[e], ti = ei[E + e];
    float x = lrelu(ssrc[(size_t)si * NH + h] + strg[(size_t)ti * NH + h]);
    float attn = expf(x - gm) / (denom[(size_t)ti * NH + h] + 1e-16f);
    atomicAdd(&out[(size_t)ti * D + t], proj[(size_t)si * D + t] * attn);
}

extern "C" void kernel_launch(void* const* d_in, const int* in_sizes, int n_in,
                              void* d_out, int out_size, void* d_ws, size_t ws_size,
                              hipStream_t stream) {
    const float* trg  = (const float*)d_in[0];
    const float* src  = (const float*)d_in[1];
    const int*   ei   = (const int*)d_in[2];
    const float* Wtrg = (const float*)d_in[3];
    const float* Wsrc = (const float*)d_in[4];
    const float* asrc = (const float*)d_in[5];
    const float* atrg = (const float*)d_in[6];
    float* out = (float*)d_out;

    const int N = in_sizes[0] / D;   // B = 1
    const int E = in_sizes[2] / 2;

    char* ws = (char*)d_ws;
    float* proj  = (float*)ws;  ws += (size_t)N * D * sizeof(float);
    float* ssrc  = (float*)ws;  ws += (size_t)N * NH * sizeof(float);
    float* strg  = (float*)ws;  ws += (size_t)N * NH * sizeof(float);
    float* denom = (float*)ws;  ws += (size_t)N * NH * sizeof(float);
    float* wtT   = (float*)ws;  ws += (size_t)NH * D * sizeof(float);
    unsigned* gmaxk = (unsigned*)ws;

    const int n_out = N * D, n_den = N * NH;

    gat_init_kernel<<<(n_out + 255) / 256, 256, 0, stream>>>(out, denom, gmaxk, n_out, n_den);
    gat_combine_a_kernel<<<(NH * D + 255) / 256, 256, 0, stream>>>(Wtrg, atrg, wtT);
    gat_src_proj_kernel<<<(N + 15) / 16, 256, 0, stream>>>(src, Wsrc, asrc, proj, ssrc, N);
    gat_trg_logits_kernel<<<(N * NH + 255) / 256, 256, 0, stream>>>(trg, wtT, strg, N);
    gat_edge_max_kernel<<<(E + 255) / 256, 256, 0, stream>>>(ei, ssrc, strg, gmaxk, E);
    gat_edge_denom_kernel<<<(E + 255) / 256, 256, 0, stream>>>(ei, ssrc, strg, gmaxk, denom, E);
    gat_edge_aggregate_kernel<<<(E + 1) / 2, 256, 0, stream>>>(ei, ssrc, strg, gmaxk, denom,
                                                               proj, out, E);
}